// PairEmbedding_19344532701514
// MI455X (gfx1250) — compile-verified
//
#include <hip/hip_runtime.h>
#include <math.h>

typedef __attribute__((ext_vector_type(2))) float v2f;
typedef __attribute__((ext_vector_type(4))) float v4f;
typedef __attribute__((ext_vector_type(8))) float v8f;

#define D_PAIR 288
#define D_HALF 144
#define BATCH  4
#define SEQLEN 256
#define NROW   (BATCH * SEQLEN)          // 1024 sequence positions
#define KDIM   D_HALF                    // 144 (reduction dim of each half-GEMM)
#define TILES_M (NROW / 16)              // 64
#define TILES_N (D_PAIR / 16)            // 18
#define TILES_PER_MAT (TILES_M * TILES_N) // 1152
#define TOTAL_TILES (2 * TILES_PER_MAT)   // 2304 waves, exact multiple of 8 waves/block
#define ROW4 (D_PAIR / 4)                // 72 float4 per output row
#define ELEM4_PER_BLK (SEQLEN * ROW4)    // 18432 float4 per (b,l) block

// Sinusoidal PE element: pe[p][d] = sin/cos(p / 10000^{(2*(d/2))/144})
__device__ __forceinline__ float pos_enc(int p, int d) {
    int j = d >> 1;
    // p * 10000^{-j/72} ; log(10000) = 9.210340371976184
    float arg = (float)p * __expf(-9.210340371976184f * ((float)j * (1.0f / 72.0f)));
    return (d & 1) ? __cosf(arg) : __sinf(arg);
}

// Kernel 1: tiny fp32 WMMA GEMM computing Row/Col terms (2 x 1024x288), with
// fused token gather (se = emb[seq]), bias and positional-encoding epilogue.
// One wave32 per 16x16 output tile; K loop in steps of 4 (V_WMMA_F32_16X16X4_F32).
__global__ void __launch_bounds__(256)
pair_precompute_wmma(const int* __restrict__ seq,
                     const int* __restrict__ aa_idx,
                     const float* __restrict__ emb,    // (21,144)
                     const float* __restrict__ W,      // (289,288)
                     const float* __restrict__ bias,   // (288,)
                     float* __restrict__ rowT,         // (1024,288)
                     float* __restrict__ colT)         // (1024,288)
{
    const int wave = (blockIdx.x * blockDim.x + threadIdx.x) >> 5;
    const int lane = threadIdx.x & 31;
    // grid sized exactly: no partial waves -> EXEC all-1s for every WMMA
    const int mat = wave / TILES_PER_MAT;       // 0: Row term (W rows 144..287), 1: Col term (W rows 0..143)
    const int rem = wave - mat * TILES_PER_MAT;
    const int tm  = rem / TILES_N;
    const int tn  = rem - tm * TILES_N;

    const int half = lane >> 4;                 // 0 or 1
    const int l15  = lane & 15;
    const int kb   = half * 2;                  // K sub-index base for this half
    const int aRow = tm * 16 + l15;             // A-frag: M = lane&15
    const int col  = tn * 16 + l15;             // B/C/D: N = lane&15
    const int wOff = (mat == 0) ? D_HALF : 0;   // which 144-row slab of W_proj

    const float* aBase = emb + seq[aRow] * D_HALF;  // gathered se row

    v8f acc = {};
    #pragma unroll 4
    for (int k0 = 0; k0 < KDIM; k0 += 4) {
        v2f a, b;
        // A 16x4 f32 layout: VGPR0 = K=kb (lanes 0-15 K=0, 16-31 K=2), VGPR1 = K=kb+1
        a.x = aBase[k0 + kb + 0];
        a.y = aBase[k0 + kb + 1];
        // B 4x16 f32 layout (mirrors A): row K = kb + {0,1}, col N = lane&15
        b.x = W[(wOff + k0 + kb + 0) * D_PAIR + col];
        b.y = W[(wOff + k0 + kb + 1) * D_PAIR + col];
        acc = __builtin_amdgcn_wmma_f32_16x16x4_f32(
            /*neg_a=*/false, a, /*neg_b=*/false, b,
            /*c_mod=*/(short)0, acc, /*reuse_a=*/false, /*reuse_b=*/false);
    }

    // Epilogue: C/D 16x16 f32 layout: M = (lane>>4)*8 + v, N = lane&15
    float* dst = (mat == 0) ? rowT : colT;
    #pragma unroll
    for (int v = 0; v < 8; ++v) {
        const int r = tm * 16 + half * 8 + v;   // sequence-position index (b*L+l)
        float val = acc[v];
        if (mat == 0) {
            val += bias[col];
            if (col < D_HALF) val += pos_enc(aa_idx[r], col);          // pe_row half
        } else {
            if (col >= D_HALF) val += pos_enc(aa_idx[r], col - D_HALF); // pe_col half
        }
        dst[r * D_PAIR + col] = val;
    }
}

// Kernel 2: bandwidth-bound fusion. One block per (b,l). Stages the row term
// and w_last in LDS, streams col-term rows from L2 (colT is 1.2 MB << 192 MB L2),
// and emits coalesced NON-TEMPORAL 128-bit stores of out[b,l,:,:] (302 MB total,
// write-once/never-re-read -> NT hint keeps the hot colT/rowT set resident in L2;
// HBM floor at 23.3 TB/s is ~13 us and this kernel is pure store bandwidth).
__global__ void __launch_bounds__(256)
pair_fuse_write(const int* __restrict__ aa_idx,
                const float* __restrict__ rowT,
                const float* __restrict__ colT,
                const float* __restrict__ wlast,   // W_proj row 288
                float* __restrict__ out)
{
    __shared__ float sA[D_PAIR];
    __shared__ float sW[D_PAIR];
    __shared__ int   sAi[SEQLEN];

    const int blk = blockIdx.x;        // = b*SEQLEN + l
    const int b   = blk >> 8;
    const int tid = threadIdx.x;

    for (int i = tid; i < D_PAIR; i += 256) {
        sA[i] = rowT[blk * D_PAIR + i];
        sW[i] = wlast[i];
    }
    for (int i = tid; i < SEQLEN; i += 256) sAi[i] = aa_idx[b * SEQLEN + i];
    __syncthreads();

    const int ai_l = aa_idx[blk];

    const v4f* colT4 = (const v4f*)(colT + (size_t)b * SEQLEN * D_PAIR);
    v4f*       out4  = (v4f*)(out + (size_t)blk * SEQLEN * D_PAIR);
    const v4f* sA4   = (const v4f*)sA;
    const v4f* sW4   = (const v4f*)sW;

    for (int e = tid; e < ELEM4_PER_BLK; e += 256) {
        const int k = e / ROW4;            // const divisor -> mul/shift
        const int q = e - k * ROW4;
        const float sep = __logf(fabsf((float)(ai_l - sAi[k])) + 1.0f);
        const v4f a = sA4[q];
        const v4f c = colT4[k * ROW4 + q];   // regular-temporal: L2-resident, reused 256x
        const v4f w = sW4[q];
        v4f o = a + c + sep * w;             // element-wise on native vector
        __builtin_nontemporal_store(o, &out4[e]);  // global_store_b128 th:TH_STORE_NT
    }
}

extern "C" void kernel_launch(void* const* d_in, const int* in_sizes, int n_in,
                              void* d_out, int out_size, void* d_ws, size_t ws_size,
                              hipStream_t stream) {
    const int*   seq    = (const int*)d_in[0];   // (4,256) int32
    const int*   aa_idx = (const int*)d_in[1];   // (4,256) int32
    const float* emb    = (const float*)d_in[2]; // (21,144) f32
    const float* W      = (const float*)d_in[3]; // (289,288) f32
    const float* bias   = (const float*)d_in[4]; // (288,) f32
    float*       out    = (float*)d_out;         // (4,256,256,288) f32

    float* rowT = (float*)d_ws;                  // 1024*288 f32
    float* colT = rowT + NROW * D_PAIR;          // 1024*288 f32  (total 2.36 MB)

    // 2304 waves * 32 lanes / 256 threads = 288 blocks, exact (EXEC all-1s)
    pair_precompute_wmma<<<(TOTAL_TILES * 32) / 256, 256, 0, stream>>>(
        seq, aa_idx, emb, W, bias, rowT, colT);

    pair_fuse_write<<<NROW, 256, 0, stream>>>(
        aa_idx, rowT, colT, W + D_PAIR * D_PAIR, out);
}